// RNARefiner_8727373545932
// MI455X (gfx1250) — compile-verified
//
#include <hip/hip_runtime.h>
#include <math.h>

typedef __attribute__((ext_vector_type(2))) float v2f;
typedef __attribute__((ext_vector_type(8))) float v8f;

#define B_ 2
#define L_ 512
#define D_ 256
#define H_ 64
#define S_ 4

__device__ __forceinline__ float gelu_exact(float x) {
    return 0.5f * x * (1.0f + erff(x * 0.7071067811865475f));
}

// ---------------------------------------------------------------------------
// Kernel 0: build local frames R (B,L,3,3 row-major) and t (B,L,3) from coords
// ---------------------------------------------------------------------------
__global__ __launch_bounds__(256)
void k_frames(const float* __restrict__ coords, const unsigned char* __restrict__ mask,
              float* __restrict__ Rw, float* __restrict__ tw)
{
    int idx = blockIdx.x * blockDim.x + threadIdx.x;
    if (idx >= B_ * L_) return;
    const float* c = coords + idx * 9;
    float P0 = c[0], P1 = c[1], P2 = c[2];
    float C0 = c[3], C1 = c[4], C2 = c[5];
    float N0 = c[6], N1 = c[7], N2 = c[8];

    float v0 = N0 - C0, v1 = N1 - C1, v2 = N2 - C2;
    float n1 = fmaxf(sqrtf(v0*v0 + v1*v1 + v2*v2), 1e-12f);
    float e1x = v0/n1, e1y = v1/n1, e1z = v2/n1;

    float u0 = P0 - C0, u1 = P1 - C1, u2 = P2 - C2;
    float dp = u0*e1x + u1*e1y + u2*e1z;
    float w0 = u0 - dp*e1x, w1 = u1 - dp*e1y, w2v = u2 - dp*e1z;
    float n2 = fmaxf(sqrtf(w0*w0 + w1*w1 + w2v*w2v), 1e-12f);
    float e2x = w0/n2, e2y = w1/n2, e2z = w2v/n2;

    float e3x = e1y*e2z - e1z*e2y;
    float e3y = e1z*e2x - e1x*e2z;
    float e3z = e1x*e2y - e1y*e2x;

    bool m = mask[idx] != 0;
    float e1a[3] = {e1x, e1y, e1z};
    float e2a[3] = {e2x, e2y, e2z};
    float e3a[3] = {e3x, e3y, e3z};
    #pragma unroll
    for (int x = 0; x < 3; ++x) {
        Rw[idx*9 + x*3 + 0] = m ? e1a[x] : (x == 0 ? 1.f : 0.f);
        Rw[idx*9 + x*3 + 1] = m ? e2a[x] : (x == 1 ? 1.f : 0.f);
        Rw[idx*9 + x*3 + 2] = m ? e3a[x] : (x == 2 ? 1.f : 0.f);
    }
    tw[idx*3 + 0] = C0;
    tw[idx*3 + 1] = C1;
    tw[idx*3 + 2] = C2;
}

// ---------------------------------------------------------------------------
// Kernel 1 (O(L^2), WMMA): invariants + pair layer-1 (7->64) + gelu + masked
// sum over j  ->  xsum[B,L,64].
// WMMA f32 16x16x4: M = 16 j's, K = 7 padded to 2 chunks of 4, N = 64 (4 tiles).
// Block = 256 thr (8 waves) handles 16 i's; per-batch-row R/t/mask live in LDS.
// ---------------------------------------------------------------------------
__global__ __launch_bounds__(256)
void k_pair_wmma(const float* __restrict__ Rw, const float* __restrict__ tw,
                 const unsigned char* __restrict__ mask,
                 const float* __restrict__ w1s, const float* __restrict__ b1s,
                 float* __restrict__ xsum)
{
    __shared__ float sR[L_ * 9];
    __shared__ float st[L_ * 3];
    __shared__ float smask[L_];
    __shared__ float sw1[7 * H_];
    __shared__ float sb1[H_];

    const int b  = blockIdx.y;
    const int i0 = blockIdx.x * 16;
    const int tid = threadIdx.x;

    for (int j = tid; j < L_; j += 256) {
        smask[j] = mask[b*L_ + j] ? 1.0f : 0.0f;
        #pragma unroll
        for (int q = 0; q < 3; ++q) st[j*3 + q] = tw[(b*L_ + j)*3 + q];
        #pragma unroll
        for (int q = 0; q < 9; ++q) sR[j*9 + q] = Rw[(b*L_ + j)*9 + q];
    }
    if (tid < 7 * H_) sw1[tid] = w1s[tid];
    if (tid < H_)     sb1[tid] = b1s[tid];
    __syncthreads();

    const int wave = tid >> 5;
    const int lane = tid & 31;
    const int lo   = lane & 15;
    const bool low = lane < 16;

    // B-matrix regs (4x16 f32 per ISA layout): constant across the block.
    v2f bm0[4], bm1[4];
    float bias[4];
    #pragma unroll
    for (int nt = 0; nt < 4; ++nt) {
        int N  = nt * 16 + lo;
        int k0 = low ? 0 : 2;
        bm0[nt].x = sw1[k0*H_ + N];
        bm0[nt].y = sw1[(k0+1)*H_ + N];
        if (low) { bm1[nt].x = sw1[4*H_ + N]; bm1[nt].y = sw1[5*H_ + N]; }
        else     { bm1[nt].x = sw1[6*H_ + N]; bm1[nt].y = 0.0f; }   // K=7 pad
        bias[nt] = sb1[N];
    }

    for (int ii = 0; ii < 2; ++ii) {
        const int i = i0 + wave*2 + ii;
        float Ri[9];
        #pragma unroll
        for (int q = 0; q < 9; ++q) Ri[q] = sR[i*9 + q];
        const float ti0 = st[i*3+0], ti1 = st[i*3+1], ti2 = st[i*3+2];
        const float mi = smask[i];

        float acc[4] = {0.f, 0.f, 0.f, 0.f};

        for (int j0 = 0; j0 < L_; j0 += 16) {
            const int j = j0 + lo;
            // --- pair invariants (i, j) ---
            float dx = st[j*3+0] - ti0, dy = st[j*3+1] - ti1, dz = st[j*3+2] - ti2;
            float dl0 = Ri[0]*dx + Ri[3]*dy + Ri[6]*dz;   // R_i^T * dt
            float dl1 = Ri[1]*dx + Ri[4]*dy + Ri[7]*dz;
            float dl2 = Ri[2]*dx + Ri[5]*dy + Ri[8]*dz;
            float dist = fmaxf(sqrtf(dx*dx + dy*dy + dz*dz), 1e-4f);
            float logd = logf(dist);
            float tr = 0.f;
            #pragma unroll
            for (int q = 0; q < 9; ++q) tr += Ri[q] * sR[j*9 + q];
            float cosa = fminf(fmaxf((tr - 1.0f) * 0.5f, -1.0f), 1.0f);
            float inv0 = dist, inv1 = dl0, inv2 = dl1, inv3 = dl2;
            float inv4 = tr, inv5 = cosa, inv6 = logd;

            // A-matrix (16x4 f32): lanes 0-15 hold K{0,1}/{4,5}, 16-31 hold K{2,3}/{6,pad}
            v2f a0, a1;
            if (low) { a0.x = inv0; a0.y = inv1; a1.x = inv4; a1.y = inv5; }
            else     { a0.x = inv2; a0.y = inv3; a1.x = inv6; a1.y = 0.0f; }

            #pragma unroll
            for (int nt = 0; nt < 4; ++nt) {
                v8f cacc = {bias[nt], bias[nt], bias[nt], bias[nt],
                            bias[nt], bias[nt], bias[nt], bias[nt]};
                cacc = __builtin_amdgcn_wmma_f32_16x16x4_f32(
                    false, a0, false, bm0[nt], (short)0, cacc, false, false);
                cacc = __builtin_amdgcn_wmma_f32_16x16x4_f32(
                    false, a1, false, bm1[nt], (short)0, cacc, false, false);
                // gelu + mask + reduce over rows (M = j within tile)
                #pragma unroll
                for (int r = 0; r < 8; ++r) {
                    int jr = j0 + r + (low ? 0 : 8);
                    acc[nt] += gelu_exact(cacc[r]) * mi * smask[jr];
                }
            }
        }
        // combine lane l (rows 0-7) with lane l+16 (rows 8-15), store 64 features
        #pragma unroll
        for (int nt = 0; nt < 4; ++nt) {
            float other = __shfl_down(acc[nt], 16, 32);
            if (low) xsum[(b*L_ + i)*H_ + nt*16 + lo] = acc[nt] + other;
        }
    }
}

// ---------------------------------------------------------------------------
// Kernel 2: per (b,i): agg = xsum@w2/n + b2; LN(h); gate; h_aug; FFN -> 6;
// Rodrigues frame update; reconstruct atoms; write outputs.
// ---------------------------------------------------------------------------
__device__ __forceinline__ float block_reduce_sum(float v, float* red, int tid) {
    red[tid] = v; __syncthreads();
    for (int s = 128; s > 0; s >>= 1) {
        if (tid < s) red[tid] += red[tid + s];
        __syncthreads();
    }
    float r = red[0]; __syncthreads();
    return r;
}

__global__ __launch_bounds__(256)
void k_row_update(const float* __restrict__ h, const unsigned char* __restrict__ mask,
                  const float* __restrict__ xsum,
                  float* __restrict__ Rw, float* __restrict__ tw,
                  const float* __restrict__ w2s, const float* __restrict__ b2s,
                  const float* __restrict__ gw_ws, const float* __restrict__ gw_bs,
                  const float* __restrict__ ga_ws, const float* __restrict__ ga_bs,
                  const float* __restrict__ ln_gs, const float* __restrict__ ln_bs,
                  const float* __restrict__ fh_ln_g, const float* __restrict__ fh_ln_b,
                  const float* __restrict__ fh_w1, const float* __restrict__ fh_b1,
                  const float* __restrict__ fh_w2, const float* __restrict__ fh_b2,
                  float* __restrict__ out_last, float* __restrict__ out_stack,
                  int write_last)
{
    __shared__ float cat[2 * D_];
    __shared__ float aux[D_];
    __shared__ float red[256];
    __shared__ float xs[H_];
    __shared__ float fv[D_ / 2];
    __shared__ float fh6[8];

    const int bi = blockIdx.x;         // b*L + i
    const int b  = bi / L_;
    const int tid = threadIdx.x;

    // n_v = max(sum_j pm_ij, 1); pm_ij = mask_i & mask_j
    float msum = (mask[b*L_ + tid] ? 1.f : 0.f) + (mask[b*L_ + tid + 256] ? 1.f : 0.f);
    float cnt = block_reduce_sum(msum, red, tid);
    const float mi = mask[bi] ? 1.f : 0.f;
    const float cntrow = mi * cnt;
    const float n_v = fmaxf(cntrow, 1.0f);

    if (tid < H_) xs[tid] = xsum[bi*H_ + tid];
    __syncthreads();

    // agg = (xsum @ w2 + cntrow*b2) / n_v     (w2 hoisted out of pair loop)
    float aggv = cntrow * b2s[tid];
    #pragma unroll 4
    for (int k = 0; k < H_; ++k) aggv += xs[k] * w2s[k*D_ + tid];
    aggv /= n_v;

    // layer norm of h
    float hv = h[bi*D_ + tid];
    float mu = block_reduce_sum(hv, red, tid) * (1.0f / D_);
    float dv = hv - mu;
    float var = block_reduce_sum(dv*dv, red, tid) * (1.0f / D_);
    float hs = dv * rsqrtf(var + 1e-5f) * ln_gs[tid] + ln_bs[tid];

    cat[tid]      = hs;
    cat[D_ + tid] = aggv;
    __syncthreads();

    // g / a GEMVs from LDS-resident cat (coalesced weight reads, stride D over k)
    float gv = gw_bs[tid], av = ga_bs[tid];
    for (int k = 0; k < 2*D_; ++k) {
        float cv = cat[k];
        gv += cv * gw_ws[k*D_ + tid];
        av += cv * ga_ws[k*D_ + tid];
    }
    av = 1.0f / (1.0f + expf(-av));
    float haug = hv + av * gv;
    aux[tid] = haug;
    __syncthreads();

    // layer norm of h_aug
    float mu2 = block_reduce_sum(haug, red, tid) * (1.0f / D_);
    float d2 = haug - mu2;
    float var2 = block_reduce_sum(d2*d2, red, tid) * (1.0f / D_);
    float hn = d2 * rsqrtf(var2 + 1e-5f) * fh_ln_g[tid] + fh_ln_b[tid];
    __syncthreads();
    cat[tid] = hn;
    __syncthreads();

    // FFN 256 -> 128 -> 6
    if (tid < 128) {
        float u = fh_b1[tid];
        for (int k = 0; k < D_; ++k) u += cat[k] * fh_w1[k*128 + tid];
        fv[tid] = gelu_exact(u);
    }
    __syncthreads();
    if (tid < 6) {
        float u = fh_b2[tid];
        for (int k = 0; k < 128; ++k) u += fv[k] * fh_w2[k*6 + tid];
        fh6[tid] = u;
    }
    __syncthreads();

    if (tid == 0) {
        float dr0 = fh6[0]*mi, dr1 = fh6[1]*mi, dr2 = fh6[2]*mi;
        float dt0 = fh6[3]*mi, dt1 = fh6[4]*mi, dt2 = fh6[5]*mi;
        float ang = fmaxf(sqrtf(dr0*dr0 + dr1*dr1 + dr2*dr2), 1e-8f);
        float inva = 1.0f / ang;
        float ax = dr0*inva, ay = dr1*inva, az = dr2*inva;
        float ca = cosf(ang), sa = sinf(ang), omc = 1.0f - ca;
        float Rd[9];
        Rd[0] = ca + omc*ax*ax;      Rd[1] = -sa*az + omc*ax*ay;  Rd[2] =  sa*ay + omc*ax*az;
        Rd[3] =  sa*az + omc*ay*ax;  Rd[4] = ca + omc*ay*ay;      Rd[5] = -sa*ax + omc*ay*az;
        Rd[6] = -sa*ay + omc*az*ax;  Rd[7] =  sa*ax + omc*az*ay;  Rd[8] = ca + omc*az*az;

        float Ro[9], to[3];
        #pragma unroll
        for (int q = 0; q < 9; ++q) Ro[q] = Rw[bi*9 + q];
        #pragma unroll
        for (int q = 0; q < 3; ++q) to[q] = tw[bi*3 + q];

        float Rn[9];
        #pragma unroll
        for (int x = 0; x < 3; ++x)
            #pragma unroll
            for (int y = 0; y < 3; ++y) {
                float sacc = 0.f;
                #pragma unroll
                for (int z = 0; z < 3; ++z) sacc += Rd[x*3 + z] * Ro[z*3 + y];
                Rn[x*3 + y] = sacc;
            }
        float tn[3];
        tn[0] = to[0] + Ro[0]*dt0 + Ro[1]*dt1 + Ro[2]*dt2;
        tn[1] = to[1] + Ro[3]*dt0 + Ro[4]*dt1 + Ro[5]*dt2;
        tn[2] = to[2] + Ro[6]*dt0 + Ro[7]*dt1 + Ro[8]*dt2;

        #pragma unroll
        for (int q = 0; q < 9; ++q) Rw[bi*9 + q] = Rn[q];
        #pragma unroll
        for (int q = 0; q < 3; ++q) tw[bi*3 + q] = tn[q];

        const float T[3][3] = {{-0.9f, 1.2f, 0.f}, {0.f, 0.f, 0.f}, {2.5f, 0.f, 0.f}};
        #pragma unroll
        for (int a = 0; a < 3; ++a)
            #pragma unroll
            for (int x = 0; x < 3; ++x) {
                float cc = tn[x] + Rn[x*3+0]*T[a][0] + Rn[x*3+1]*T[a][1] + Rn[x*3+2]*T[a][2];
                int off = (bi*3 + a)*3 + x;
                out_stack[off] = cc;
                if (write_last) out_last[off] = cc;
            }
    }
}

// ---------------------------------------------------------------------------
extern "C" void kernel_launch(void* const* d_in, const int* in_sizes, int n_in,
                              void* d_out, int out_size, void* d_ws, size_t ws_size,
                              hipStream_t stream)
{
    (void)in_sizes; (void)n_in; (void)out_size; (void)ws_size;

    const float* h        = (const float*)d_in[0];
    const float* coords   = (const float*)d_in[1];
    const unsigned char* mask = (const unsigned char*)d_in[2];
    const float* pair_w1  = (const float*)d_in[3];
    const float* pair_b1  = (const float*)d_in[4];
    const float* pair_w2  = (const float*)d_in[5];
    const float* pair_b2  = (const float*)d_in[6];
    const float* gw_w     = (const float*)d_in[7];
    const float* gw_b     = (const float*)d_in[8];
    const float* ga_w     = (const float*)d_in[9];
    const float* ga_b     = (const float*)d_in[10];
    const float* ln_g     = (const float*)d_in[11];
    const float* ln_b     = (const float*)d_in[12];
    const float* fh_ln_g  = (const float*)d_in[13];
    const float* fh_ln_b  = (const float*)d_in[14];
    const float* fh_w1    = (const float*)d_in[15];
    const float* fh_b1    = (const float*)d_in[16];
    const float* fh_w2    = (const float*)d_in[17];
    const float* fh_b2    = (const float*)d_in[18];

    float* Rw   = (float*)d_ws;            // B*L*9
    float* tw   = Rw + B_*L_*9;            // B*L*3
    float* xsum = tw + B_*L_*3;            // B*L*64

    float* out_last  = (float*)d_out;            // B*L*3*3
    float* out_stack = out_last + B_*L_*9;       // S*B*L*3*3

    k_frames<<<(B_*L_ + 255)/256, 256, 0, stream>>>(coords, mask, Rw, tw);

    for (int s = 0; s < S_; ++s) {
        k_pair_wmma<<<dim3(L_/16, B_), 256, 0, stream>>>(
            Rw, tw, mask, pair_w1 + s*7*H_, pair_b1 + s*H_, xsum);

        k_row_update<<<B_*L_, 256, 0, stream>>>(
            h, mask, xsum, Rw, tw,
            pair_w2 + s*H_*D_, pair_b2 + s*D_,
            gw_w + s*2*D_*D_, gw_b + s*D_,
            ga_w + s*2*D_*D_, ga_b + s*D_,
            ln_g + s*D_, ln_b + s*D_,
            fh_ln_g, fh_ln_b, fh_w1, fh_b1, fh_w2, fh_b2,
            out_last, out_stack + s*B_*L_*9, (s == S_-1) ? 1 : 0);
    }
}